// CyberBrainV6_51539608083
// MI455X (gfx1250) — compile-verified
//
#include <hip/hip_runtime.h>
#include <hip/hip_bf16.h>

// ---------------------------------------------------------------------------
// CyberBrainV6 for MI455X (gfx1250, wave32).
// Memory-bound pipeline (~1.8 GiB HBM traffic -> ~77us floor @ 23.3 TB/s).
// GEMMs (expert FFN, lm_head) use V_WMMA_F32_16X16X4_F32 to keep f32 math.
// ---------------------------------------------------------------------------

#define H_DIM 1024
#define V_DIM 15000
#define NLAYER 2
#define NEXP 4
#define BB 32
#define SS 2048

typedef __attribute__((ext_vector_type(2))) float v2f;
typedef __attribute__((ext_vector_type(8))) float v8f;

// ---------------------------------------------------------------------------
// Kernel 1: x[tk,:] = emb[windows[tk],:]  and  rstd[tk] = rsqrt(mean(x^2)+eps)
// One 256-thread block per token (B*S = 65536 blocks), float4 per thread.
// ---------------------------------------------------------------------------
__global__ void gather_rstd_kernel(const int* __restrict__ windows,
                                   const float* __restrict__ emb,
                                   float* __restrict__ x,
                                   float* __restrict__ rstd) {
  __shared__ float red[256];
  const int tk = blockIdx.x;
  const int tid = threadIdx.x;
  const int wid = windows[tk];
  const float4 v = ((const float4*)(emb + (size_t)wid * H_DIM))[tid];
  ((float4*)(x + (size_t)tk * H_DIM))[tid] = v;
  red[tid] = v.x * v.x + v.y * v.y + v.z * v.z + v.w * v.w;
  __syncthreads();
  for (int off = 128; off > 0; off >>= 1) {
    if (tid < off) red[tid] += red[tid + off];
    __syncthreads();
  }
  if (tid == 0) rstd[tk] = rsqrtf(red[0] * (1.0f / H_DIM) + 1e-6f);
}

// ---------------------------------------------------------------------------
// Kernel 2: rstd[tk] over (x[tk,:] + add[b,:])  (layer-1 path; add folded in)
// ---------------------------------------------------------------------------
__global__ void rstd_add_kernel(const float* __restrict__ x,
                                const float* __restrict__ add,
                                float* __restrict__ rstd) {
  __shared__ float red[256];
  const int tk = blockIdx.x;
  const int b = tk / SS;
  const int tid = threadIdx.x;
  float4 v = ((const float4*)(x + (size_t)tk * H_DIM))[tid];
  const float4 a = ((const float4*)(add + (size_t)b * H_DIM))[tid];
  v.x += a.x; v.y += a.y; v.z += a.z; v.w += a.w;
  red[tid] = v.x * v.x + v.y * v.y + v.z * v.z + v.w * v.w;
  __syncthreads();
  for (int off = 128; off > 0; off >>= 1) {
    if (tid < off) red[tid] += red[tid + off];
    __syncthreads();
  }
  if (tid == 0) rstd[tk] = rsqrtf(red[0] * (1.0f / H_DIM) + 1e-6f);
}

// ---------------------------------------------------------------------------
// Kernel 3: EMA scan along S, in place.  One thread per (b,h) channel
// (32768 threads, blockDim 128 -> 256 blocks; lanes cover contiguous h ->
// fully coalesced 128B/wave loads).  t unrolled x4 for memory-level parallelism.
//   xn_t = (x_t + add)*rstd_t*w1[h];  s = s*d + xn_t*(1-d);  x_t <- x_t+add+s
// ---------------------------------------------------------------------------
__global__ void scan_kernel(float* __restrict__ x,
                            const float* __restrict__ rstd,
                            const float* __restrict__ add, int has_add,
                            const float* __restrict__ norm1_w,
                            const float* __restrict__ decay_logit) {
  const int gid = blockIdx.x * blockDim.x + threadIdx.x;  // 0..B*H-1
  const int b = gid / H_DIM;
  const int h = gid % H_DIM;
  const float w = norm1_w[h];
  const float z = decay_logit[h];
  const float d = 1.0f / (1.0f + __expf(-z));
  const float omd = 1.0f - d;
  const float a = has_add ? add[(size_t)b * H_DIM + h] : 0.0f;
  const float* rs = rstd + (size_t)b * SS;
  float* xp = x + (size_t)b * SS * H_DIM + h;
  float s = 0.0f;
  for (int t = 0; t < SS; t += 4) {
    float v0 = xp[0 * H_DIM];
    float v1 = xp[1 * H_DIM];
    float v2 = xp[2 * H_DIM];
    float v3 = xp[3 * H_DIM];
    const float r0 = rs[t + 0], r1 = rs[t + 1], r2 = rs[t + 2], r3 = rs[t + 3];
    v0 += a; s = s * d + (v0 * r0 * w) * omd; xp[0 * H_DIM] = v0 + s;
    v1 += a; s = s * d + (v1 * r1 * w) * omd; xp[1 * H_DIM] = v1 + s;
    v2 += a; s = s * d + (v2 * r2 * w) * omd; xp[2 * H_DIM] = v2 + s;
    v3 += a; s = s * d + (v3 * r3 * w) * omd; xp[3 * H_DIM] = v3 + s;
    xp += 4 * H_DIM;
  }
}

// ---------------------------------------------------------------------------
// Kernel 4: rownorm of the LAST token: out[b,:] = rmsnorm(x[b,S-1,:]+add)*w
// One block per b (32 blocks, 256 threads, float4).
// ---------------------------------------------------------------------------
__global__ void rownorm_kernel(const float* __restrict__ x,
                               const float* __restrict__ add, int has_add,
                               const float* __restrict__ w,
                               float* __restrict__ out) {
  __shared__ float red[256];
  __shared__ float srstd;
  const int b = blockIdx.x;
  const int tid = threadIdx.x;
  float4 v = ((const float4*)(x + ((size_t)b * SS + (SS - 1)) * H_DIM))[tid];
  if (has_add) {
    const float4 a = ((const float4*)(add + (size_t)b * H_DIM))[tid];
    v.x += a.x; v.y += a.y; v.z += a.z; v.w += a.w;
  }
  red[tid] = v.x * v.x + v.y * v.y + v.z * v.z + v.w * v.w;
  __syncthreads();
  for (int off = 128; off > 0; off >>= 1) {
    if (tid < off) red[tid] += red[tid + off];
    __syncthreads();
  }
  if (tid == 0) srstd = rsqrtf(red[0] * (1.0f / H_DIM) + 1e-6f);
  __syncthreads();
  const float r = srstd;
  const float4 wv = ((const float4*)w)[tid];
  float4 o;
  o.x = v.x * r * wv.x; o.y = v.y * r * wv.y;
  o.z = v.z * r * wv.z; o.w = v.w * r * wv.w;
  ((float4*)(out + (size_t)b * H_DIM))[tid] = o;
}

// ---------------------------------------------------------------------------
// Kernel 5: expert FFN via WMMA f32 16x16x4.
//   out_all[e] = relu(pool @ Wexp[l,e]^T); epilogue selects experts[b]==e and
//   writes add_out[b,:].  One wave per (e, mtile, ntile): 4*2*64 = 512 waves.
// A/B lane layout (ISA 7.12.2): elem(row = lane%16, k = 2*(lane/16)+v).
// C/D: (M = v + 8*(lane/16), N = lane%16).
// ---------------------------------------------------------------------------
__global__ void expert_gemm_kernel(const float* __restrict__ pool,
                                   const float* __restrict__ Wexp_l,
                                   const int* __restrict__ experts,
                                   float* __restrict__ add_out) {
  const int NT = H_DIM / 16;  // 64
  const int bid = blockIdx.x;
  const int e = bid / (2 * NT);
  const int rem = bid % (2 * NT);
  const int mt = rem / NT;
  const int nt = rem % NT;
  const int lane = threadIdx.x;
  const int r = lane & 15;
  const int kk = (lane >> 4) << 1;  // 0 or 2
  const float* arow = pool + (size_t)(mt * 16 + r) * H_DIM;
  const float* brow = Wexp_l + ((size_t)e * H_DIM + (nt * 16 + r)) * H_DIM;
  v8f c = {};
  for (int k0 = 0; k0 < H_DIM; k0 += 4) {
    v2f a = *(const v2f*)(arow + k0 + kk);
    v2f bm = *(const v2f*)(brow + k0 + kk);
    c = __builtin_amdgcn_wmma_f32_16x16x4_f32(false, a, false, bm,
                                              (short)0, c, false, false);
  }
  const int nbase = nt * 16 + r;
  const int moff = (lane >> 4) << 3;  // +8 for upper half-wave
#pragma unroll
  for (int v = 0; v < 8; ++v) {
    const int brw = mt * 16 + v + moff;
    if (experts[brw] == e) {
      const float val = c[v];
      add_out[(size_t)brw * H_DIM + nbase] = val > 0.0f ? val : 0.0f;
    }
  }
}

// ---------------------------------------------------------------------------
// Kernel 6: logits = fs @ lm_head^T via WMMA f32 16x16x4.
// One wave per (mtile, ntile): 2 * ceil(15000/16)=938 -> 1876 waves.
// Edge column tile masked branch-free (EXEC all-1s at the WMMA).
// ---------------------------------------------------------------------------
__global__ void lmhead_gemm_kernel(const float* __restrict__ fs,
                                   const float* __restrict__ lm,
                                   float* __restrict__ out) {
  const int NT = (V_DIM + 15) / 16;  // 938
  const int bid = blockIdx.x;
  const int mt = bid / NT;
  const int nt = bid % NT;
  const int lane = threadIdx.x;
  const int r = lane & 15;
  const int kk = (lane >> 4) << 1;
  const float* arow = fs + (size_t)(mt * 16 + r) * H_DIM;
  const int vrow = nt * 16 + r;
  const float msk = (vrow < V_DIM) ? 1.0f : 0.0f;
  const float* brow = lm + (size_t)(vrow < V_DIM ? vrow : 0) * H_DIM;
  v8f c = {};
  for (int k0 = 0; k0 < H_DIM; k0 += 4) {
    v2f a = *(const v2f*)(arow + k0 + kk);
    v2f bm = *(const v2f*)(brow + k0 + kk);
    bm.x *= msk; bm.y *= msk;
    c = __builtin_amdgcn_wmma_f32_16x16x4_f32(false, a, false, bm,
                                              (short)0, c, false, false);
  }
  const int col = nt * 16 + r;  // N = lane%16
  const int moff = (lane >> 4) << 3;
  if (col < V_DIM) {
#pragma unroll
    for (int v = 0; v < 8; ++v) {
      const int brw = mt * 16 + v + moff;
      out[(size_t)brw * V_DIM + col] = c[v];
    }
  }
}

// ---------------------------------------------------------------------------
// Host-side orchestration.
// Inputs: 0 windows[B,S] i32, 1 hemis (unused), 2 experts[B] i32, 3 emb[V,H],
// 4 norm1_w[L,H], 5 decay_logit[L,H], 6 norm2_w[L,H], 7 Wexp[L,E,H,H],
// 8 final_norm_w[H], 9 lm_head[V,H].  Output: logits[B,V] f32.
// ---------------------------------------------------------------------------
extern "C" void kernel_launch(void* const* d_in, const int* in_sizes, int n_in,
                              void* d_out, int out_size, void* d_ws, size_t ws_size,
                              hipStream_t stream) {
  const int* windows = (const int*)d_in[0];
  const int* experts = (const int*)d_in[2];
  const float* emb = (const float*)d_in[3];
  const float* norm1_w = (const float*)d_in[4];
  const float* decay_logit = (const float*)d_in[5];
  const float* norm2_w = (const float*)d_in[6];
  const float* Wexp = (const float*)d_in[7];
  const float* final_norm_w = (const float*)d_in[8];
  const float* lm_head = (const float*)d_in[9];
  float* out = (float*)d_out;

  float* ws = (float*)d_ws;
  float* x = ws;                                   // B*S*H   = 67,108,864 f
  float* rstd = x + (size_t)BB * SS * H_DIM;       // B*S     = 65,536 f
  float* pool = rstd + (size_t)BB * SS;            // B*H     = 32,768 f
  float* addA = pool + (size_t)BB * H_DIM;         // B*H
  float* addB = addA + (size_t)BB * H_DIM;         // B*H

  const int n_tok = BB * SS;
  const int scan_blocks = (BB * H_DIM) / 128;
  const int exp_blocks = NEXP * 2 * (H_DIM / 16);        // 512
  const int lm_blocks = 2 * ((V_DIM + 15) / 16);         // 1876
  const size_t wexp_layer = (size_t)NEXP * H_DIM * H_DIM;

  // ---- layer 0 ----
  gather_rstd_kernel<<<n_tok, 256, 0, stream>>>(windows, emb, x, rstd);
  scan_kernel<<<scan_blocks, 128, 0, stream>>>(x, rstd, nullptr, 0,
                                               norm1_w, decay_logit);
  rownorm_kernel<<<BB, 256, 0, stream>>>(x, nullptr, 0, norm2_w, pool);
  expert_gemm_kernel<<<exp_blocks, 32, 0, stream>>>(pool, Wexp, experts, addA);

  // ---- layer 1 (expert output of layer 0 folded in via addA) ----
  rstd_add_kernel<<<n_tok, 256, 0, stream>>>(x, addA, rstd);
  scan_kernel<<<scan_blocks, 128, 0, stream>>>(x, rstd, addA, 1,
                                               norm1_w + H_DIM,
                                               decay_logit + H_DIM);
  rownorm_kernel<<<BB, 256, 0, stream>>>(x, nullptr, 0, norm2_w + H_DIM, pool);
  expert_gemm_kernel<<<exp_blocks, 32, 0, stream>>>(pool, Wexp + wexp_layer,
                                                    experts, addB);

  // ---- final norm + lm_head ----
  rownorm_kernel<<<BB, 256, 0, stream>>>(x, addB, 1, final_norm_w, pool);
  lmhead_gemm_kernel<<<lm_blocks, 32, 0, stream>>>(pool, lm_head, out);
}